// MeshRenderer_cpu_13993003450755
// MI455X (gfx1250) — compile-verified
//
#include <hip/hip_runtime.h>
#include <hip/hip_bf16.h>
#include <cmath>

typedef float v2f __attribute__((ext_vector_type(2)));
typedef float v8f __attribute__((ext_vector_type(8)));

#define RSIZE 224
#define NEG_CLIP -999999.0f

// ---------------------------------------------------------------------------
// Kernel 1: vertex transform -> screen-space sx, sy and depth sz
// sx = k*vx/vz * 112 + 112 ; sy = -k*vy/vz * 112 + 112 ; sz = -vz
// ---------------------------------------------------------------------------
__global__ void k_vertex(const float* __restrict__ vtx, int N, float k,
                         float* __restrict__ sx, float* __restrict__ sy,
                         float* __restrict__ sz) {
    int i = blockIdx.x * blockDim.x + threadIdx.x;
    if (i >= N) return;
    float vx = vtx[i * 3 + 0];
    float vy = vtx[i * 3 + 1];
    float vz = vtx[i * 3 + 2];
    float invw = 1.0f / vz;
    sx[i] = (k * vx) * invw * 112.0f + 112.0f;
    sy[i] = (-k * vy) * invw * 112.0f + 112.0f;
    sz[i] = -vz;
}

// ---------------------------------------------------------------------------
// Kernel 2: per-triangle affine coefficients, written in WMMA A-operand
// layout.  For tile t (16 triangles), lane l<16 holds (a,b) of tri t*16+l
// (K=0,1), lane 16+l holds (c,0) (K=2,3).  8 floats per (tile,lane):
//   lane<16 : [ua,ub, va,vb, da,db, 0,0]
//   lane>=16: [uc,0,  vc,0,  dc,0,  0,0]
// ---------------------------------------------------------------------------
__global__ void k_trisetup(const long long* __restrict__ tri, int M, int Mpad,
                           const float* __restrict__ sx,
                           const float* __restrict__ sy,
                           const float* __restrict__ sz,
                           float* __restrict__ pack) {
    int m = blockIdx.x * blockDim.x + threadIdx.x;
    if (m >= Mpad) return;
    int t = m >> 4, l = m & 15;
    float* rowAB = pack + ((size_t)t * 32 + l) * 8;
    float* rowC  = pack + ((size_t)t * 32 + 16 + l) * 8;

    if (m >= M) {  // padded triangle: never inside
        rowAB[0] = 0.f; rowAB[1] = 0.f; rowAB[2] = 0.f; rowAB[3] = 0.f;
        rowAB[4] = 0.f; rowAB[5] = 0.f; rowAB[6] = 0.f; rowAB[7] = 0.f;
        rowC[0] = -1e30f; rowC[1] = 0.f; rowC[2] = -1e30f; rowC[3] = 0.f;
        rowC[4] = 0.f;    rowC[5] = 0.f; rowC[6] = 0.f;    rowC[7] = 0.f;
        return;
    }

    int t0 = (int)tri[m * 3 + 0];
    int t1 = (int)tri[m * 3 + 1];
    int t2 = (int)tri[m * 3 + 2];

    float p0x = sx[t0], p0y = sy[t0];
    float e0x = sx[t2] - p0x, e0y = sy[t2] - p0y;   // e0 = p2 - p0
    float e1x = sx[t1] - p0x, e1y = sy[t1] - p0y;   // e1 = p1 - p0

    float d00 = e0x * e0x + e0y * e0y;
    float d01 = e0x * e1x + e0y * e1y;
    float d11 = e1x * e1x + e1y * e1y;
    float den = d00 * d11 - d01 * d01;
    float inv = (den == 0.0f) ? 0.0f : 1.0f / den;

    float k0 = p0x * e0x + p0y * e0y;   // p0 . e0
    float k1 = p0x * e1x + p0y * e1y;   // p0 . e1

    // uu = (d11*d02 - d01*d12)*inv  with d02/d12 affine in (x,y)
    float ua = (d11 * e0x - d01 * e1x) * inv;
    float ub = (d11 * e0y - d01 * e1y) * inv;
    float uc = -(d11 * k0 - d01 * k1) * inv;
    // vv = (d00*d12 - d01*d02)*inv
    float va = (d00 * e1x - d01 * e0x) * inv;
    float vb = (d00 * e1y - d01 * e0y) * inv;
    float vc = -(d00 * k1 - d01 * k0) * inv;

    float z0 = sz[t0], z1 = sz[t1], z2 = sz[t2];
    float dz2 = z2 - z0, dz1 = z1 - z0;
    // pd = (1-u-v)*z0 + v*z1 + u*z2 = z0 + u*dz2 + v*dz1
    float da = ua * dz2 + va * dz1;
    float db = ub * dz2 + vb * dz1;
    float dc = z0 + uc * dz2 + vc * dz1;

    rowAB[0] = ua; rowAB[1] = ub;
    rowAB[2] = va; rowAB[3] = vb;
    rowAB[4] = da; rowAB[5] = db;
    rowAB[6] = 0.f; rowAB[7] = 0.f;

    rowC[0] = uc; rowC[1] = 0.f;
    rowC[2] = vc; rowC[3] = 0.f;
    rowC[4] = dc; rowC[5] = 0.f;
    rowC[6] = 0.f; rowC[7] = 0.f;
}

// ---------------------------------------------------------------------------
// Kernel 3: WMMA rasterizer.  One wave32 = 16 pixels (one 16-wide strip of a
// row, via both lane halves).  Loop over 16-triangle tiles; each tile does
// three V_WMMA_F32_16X16X4_F32 (u, v, depth) then a per-lane fused
// inside+argmax over the 8 triangle rows the lane owns.  Barycentrics of the
// winner are recomputed at the end from its affine coefficients.
// ---------------------------------------------------------------------------
__global__ void __launch_bounds__(256)
k_raster(const float* __restrict__ pack, int numTiles,
         const long long* __restrict__ tri, const float* __restrict__ feat,
         float* __restrict__ out) {
    const int W = RSIZE, HW = RSIZE * RSIZE;
    const int lane = threadIdx.x & 31;
    const int wave = blockIdx.x * (blockDim.x >> 5) + (threadIdx.x >> 5);

    const int ty = wave / (RSIZE / 16);     // pixel row 0..223
    const int tx = wave % (RSIZE / 16);     // 16-pixel strip 0..13
    const int px = tx * 16 + (lane & 15);
    const int py = ty;
    const bool hiHalf = lane >= 16;

    // B operand (4x16, K x N): column n = (x_n, y_n, 1, 0)
    v2f B;
    B[0] = hiHalf ? 1.0f : (float)px;
    B[1] = hiHalf ? 0.0f : (float)py;

    const float NEG_INF = -__builtin_huge_valf();
    float best = NEG_INF;
    int bi = 0x7FFFFFFF;
    const int rowOff = hiHalf ? 8 : 0;
    const v8f zero = {};

    const float* p = pack + ((size_t)lane) * 8;
    for (int t = 0; t < numTiles; ++t, p += 32 * 8) {
        float4 q0 = *(const float4*)p;          // (a_u,b_u, a_v,b_v) | (c_u,0, c_v,0)
        v2f    q1 = *(const v2f*)(p + 4);       // (a_d,b_d)          | (c_d,0)
        __builtin_prefetch(p + 32 * 8, 0, 3);   // next tile -> GL2 prefetch

        v2f Au; Au[0] = q0.x; Au[1] = q0.y;
        v2f Av; Av[0] = q0.z; Av[1] = q0.w;
        v2f Ad = q1;

        v8f Du = __builtin_amdgcn_wmma_f32_16x16x4_f32(
            false, Au, false, B, (short)0, zero, false, false);
        v8f Dv = __builtin_amdgcn_wmma_f32_16x16x4_f32(
            false, Av, false, B, (short)0, zero, false, false);
        v8f Dd = __builtin_amdgcn_wmma_f32_16x16x4_f32(
            false, Ad, false, B, (short)0, zero, false, false);

        const int ibase = t * 16 + rowOff;
#pragma unroll
        for (int r = 0; r < 8; ++r) {
            float u = Du[r], v = Dv[r], d = Dd[r];
            // fused inside + better-than-best; non-short-circuit & so all
            // four lane-mask compares co-issue and combine with s_and.
            // Strict > keeps first max; indices sweep ascending, matching
            // jnp.argmax tie-break.  Mixed strictness matches reference.
            bool g = (u >= 0.0f) & (v >= 0.0f) & ((u + v) < 1.0f) & (d > best);
            bi   = g ? (ibase + r) : bi;
            best = g ? d : best;
        }
    }

    // Merge the two lane halves (triangle rows 0-7 vs 8-15 of every tile).
    float oBest = __shfl_xor(best, 16, 32);
    int   oI    = __shfl_xor(bi, 16, 32);
    bool takeO = (oBest > best) || ((oBest == best) && (oI < bi));
    if (takeO) { best = oBest; bi = oI; }

    if (lane < 16) {
        bool valid = best > NEG_CLIP;
        int bsafe = valid ? bi : 0;

        // Recompute winner's barycentrics from its affine coefficients.
        int bt = bsafe >> 4, bl = bsafe & 15;
        const float* rAB = pack + ((size_t)bt * 32 + bl) * 8;
        const float* rC  = pack + ((size_t)bt * 32 + 16 + bl) * 8;
        float fx = (float)px, fy = (float)py;
        float bu = fmaf(rAB[0], fx, fmaf(rAB[1], fy, rC[0]));
        float bv = fmaf(rAB[2], fx, fmaf(rAB[3], fy, rC[2]));

        long long i0 = tri[bsafe * 3 + 0];
        long long i1 = tri[bsafe * 3 + 1];
        long long i2 = tri[bsafe * 3 + 2];
        float w0 = 1.0f - bu - bv, w1 = bv, w2 = bu;

        int idx = py * W + px;
        out[idx]      = (valid && bi > 0) ? 1.0f : 0.0f;       // mask (tb > 0)
        out[HW + idx] = valid ? -best : -0.0f;                 // depth = -dep
#pragma unroll
        for (int c = 0; c < 3; ++c) {
            float col = w0 * feat[i0 * 3 + c] + w1 * feat[i1 * 3 + c] +
                        w2 * feat[i2 * 3 + c];
            out[2 * HW + c * HW + idx] = valid ? col : 0.0f;   // image CHW
        }
    }
}

// ---------------------------------------------------------------------------
extern "C" void kernel_launch(void* const* d_in, const int* in_sizes, int n_in,
                              void* d_out, int out_size, void* d_ws, size_t ws_size,
                              hipStream_t stream) {
    const float*     vtx  = (const float*)d_in[0];
    const long long* tri  = (const long long*)d_in[1];
    const float*     feat = (const float*)d_in[2];
    float* out = (float*)d_out;

    int N = in_sizes[0] / 3;   // 20000 vertices
    int M = in_sizes[1] / 3;   // 40000 triangles
    int numTiles = (M + 15) / 16;
    int Mpad = numTiles * 16;

    float* ws = (float*)d_ws;
    float* sx = ws;
    float* sy = ws + N;
    float* sz = ws + 2 * (size_t)N;
    size_t packOff = ((3 * (size_t)N + 7) / 8) * 8;   // 32B-aligned
    float* pack = ws + packOff;                        // numTiles*32*8 floats

    // k = 1 / tan(radians(FOV/2)),  FOV = 12.594
    double half = 12.594 * 0.5 * 3.14159265358979323846 / 180.0;
    float kf = (float)(1.0 / tan(half));

    k_vertex<<<(N + 255) / 256, 256, 0, stream>>>(vtx, N, kf, sx, sy, sz);
    k_trisetup<<<(Mpad + 255) / 256, 256, 0, stream>>>(tri, M, Mpad, sx, sy, sz, pack);

    int waves = RSIZE * (RSIZE / 16);            // 3136 wave-tiles
    int blocks = waves / 8;                      // 8 waves (256 thr) per block
    k_raster<<<blocks, 256, 0, stream>>>(pack, numTiles, tri, feat, out);
}